// GradualStyleSwapEncoder_5892695130626
// MI455X (gfx1250) — compile-verified
//
#include <hip/hip_runtime.h>

typedef __attribute__((ext_vector_type(2))) float v2f;
typedef __attribute__((ext_vector_type(4))) float v4f;
typedef __attribute__((ext_vector_type(8))) float v8f;

#define NB   16      // batch
#define NS   4       // segments
#define NF   128     // feature channels
#define HW   128     // code spatial dim
#define NPIX (HW*HW) // 16384 pixels per (b,f) plane
#define NOUT 512
#define MROWS (NB*NS) // 64 rows of feats

// ---------------------------------------------------------------------------
// Kernel 1: pack segment masks (nearest 2x downsample of segmap) into one
// byte per pixel (bit s = segmap[b,s,0,2y,2x] != 0), and accumulate counts.
// Grid: 128 blocks (8 per batch) x 256 threads, 8 pixels/thread. Trivial cost.
// ---------------------------------------------------------------------------
__global__ void seg_mask_kernel(const int* __restrict__ segmap,
                                unsigned char* __restrict__ mask,
                                int* __restrict__ counts) {
    const int b     = blockIdx.x >> 3;          // 0..15
    const int chunk = blockIdx.x & 7;           // 0..7
    const int t     = threadIdx.x;              // 0..255
    const long segb = (long)b * NS * 256 * 256;

    int c0 = 0, c1 = 0, c2 = 0, c3 = 0;
    for (int i = 0; i < 8; ++i) {
        const int pix = chunk * 2048 + i * 256 + t;    // 0..16383
        const int y = pix >> 7, x = pix & 127;
        const long base = segb + (long)(2 * y) * 256 + 2 * x; // nearest: idx*2
        const int v0 = segmap[base];
        const int v1 = segmap[base + 65536];
        const int v2 = segmap[base + 2 * 65536];
        const int v3 = segmap[base + 3 * 65536];
        unsigned m = 0;
        if (v0) { m |= 1u; ++c0; }
        if (v1) { m |= 2u; ++c1; }
        if (v2) { m |= 4u; ++c2; }
        if (v3) { m |= 8u; ++c3; }
        mask[b * NPIX + pix] = (unsigned char)m;
    }
    atomicAdd(&counts[b * NS + 0], c0);
    atomicAdd(&counts[b * NS + 1], c1);
    atomicAdd(&counts[b * NS + 2], c2);
    atomicAdd(&counts[b * NS + 3], c3);
}

// ---------------------------------------------------------------------------
// Kernel 2: the bandwidth-bound pass. One block per (b,f) plane (2048 blocks,
// 256 threads). Streams 64 KiB of codes per block as non-temporal 16-byte
// loads (codes read exactly once -> don't pollute L2; mask stays hot in
// cache), accumulates 4 masked sums, LDS tree reduction, writes
// sums[(b*4+s)*128+f].
// ---------------------------------------------------------------------------
__global__ void masked_sum_kernel(const float* __restrict__ codes,
                                  const unsigned char* __restrict__ mask,
                                  float* __restrict__ sums) {
    const int bf = blockIdx.x;              // 0..2047
    const int b  = bf >> 7;                 // 0..15
    const int t  = threadIdx.x;

    const v4f*    cp = (const v4f*)(codes + (long)bf * NPIX);
    const uchar4* mp = (const uchar4*)(mask + (long)b * NPIX);

    float a0 = 0.f, a1 = 0.f, a2 = 0.f, a3 = 0.f;
#pragma unroll
    for (int i = 0; i < NPIX / 4 / 256; ++i) {     // 16 iterations
        const int idx = i * 256 + t;               // coalesced
        const v4f    v = __builtin_nontemporal_load(&cp[idx]);
        const uchar4 m = mp[idx];
        a0 += (m.x & 1u) ? v.x : 0.f;  a1 += (m.x & 2u) ? v.x : 0.f;
        a2 += (m.x & 4u) ? v.x : 0.f;  a3 += (m.x & 8u) ? v.x : 0.f;
        a0 += (m.y & 1u) ? v.y : 0.f;  a1 += (m.y & 2u) ? v.y : 0.f;
        a2 += (m.y & 4u) ? v.y : 0.f;  a3 += (m.y & 8u) ? v.y : 0.f;
        a0 += (m.z & 1u) ? v.z : 0.f;  a1 += (m.z & 2u) ? v.z : 0.f;
        a2 += (m.z & 4u) ? v.z : 0.f;  a3 += (m.z & 8u) ? v.z : 0.f;
        a0 += (m.w & 1u) ? v.w : 0.f;  a1 += (m.w & 2u) ? v.w : 0.f;
        a2 += (m.w & 4u) ? v.w : 0.f;  a3 += (m.w & 8u) ? v.w : 0.f;
    }

    __shared__ float red[NS][256];
    red[0][t] = a0; red[1][t] = a1; red[2][t] = a2; red[3][t] = a3;
    __syncthreads();
#pragma unroll
    for (int off = 128; off >= 1; off >>= 1) {
        if (t < off) {
            red[0][t] += red[0][t + off];
            red[1][t] += red[1][t + off];
            red[2][t] += red[2][t + off];
            red[3][t] += red[3][t + off];
        }
        __syncthreads();
    }
    if (t < NS) sums[(b * NS + t) * NF + (bf & 127)] = red[t][0];
}

// ---------------------------------------------------------------------------
// Kernel 3: feats = sums/counts (0 if empty) fused into the A-fragment load,
// then out[64,512] = feats[64,128] @ fc_w[512,128]^T + fc_b via
// V_WMMA_F32_16X16X4_F32 (full f32 precision). 128 waves: 4 M-tiles x 32
// N-tiles, 32 chained K=4 WMMAs each. EXEC all-ones (no divergence).
//
// A 16x4 f32 fragment (2 VGPRs): lane&15 = M-row, lane>>4 selects K={0,1} vs
// K={2,3}. B 4x16: same pattern over N columns. C/D (8 VGPRs): VGPR r holds
// rows r (lanes 0-15) and r+8 (lanes 16-31), N = lane&15.
// ---------------------------------------------------------------------------
__global__ void feats_gemm_wmma(const float* __restrict__ sums,
                                const int* __restrict__ counts,
                                const float* __restrict__ fcw,
                                const float* __restrict__ fcb,
                                float* __restrict__ out) {
    const int wave  = (blockIdx.x * blockDim.x + threadIdx.x) >> 5; // 0..127
    const int lane  = threadIdx.x & 31;
    const int mTile = wave >> 5;     // 0..3
    const int nTile = wave & 31;     // 0..31
    const int lr    = lane & 15;
    const int half  = lane >> 4;
    const int koff  = half * 2;

    const int m = mTile * 16 + lr;   // feats row for A fragment
    const int n = nTile * 16 + lr;   // output col for B fragment / C store

    const int   cnt   = counts[m];
    const float scale = (cnt > 0) ? (1.0f / (float)cnt) : 0.0f;
    const float* __restrict__ arow = sums + (long)m * NF;
    const float* __restrict__ brow = fcw  + (long)n * NF;

    v8f c = {};
#pragma unroll
    for (int ks = 0; ks < NF / 4; ++ks) {       // 32 WMMA steps
        const int k = ks * 4 + koff;
        v2f a, bmat;
        a.x    = arow[k]     * scale;
        a.y    = arow[k + 1] * scale;
        bmat.x = brow[k];
        bmat.y = brow[k + 1];
        c = __builtin_amdgcn_wmma_f32_16x16x4_f32(
                /*neg_a=*/false, a, /*neg_b=*/false, bmat,
                /*c_mod=*/(short)0, c, /*reuse_a=*/false, /*reuse_b=*/false);
    }

    const float bias = fcb[n];
#pragma unroll
    for (int r = 0; r < 8; ++r) {
        const int mo = mTile * 16 + r + half * 8;
        out[(long)mo * NOUT + n] = c[r] + bias;
    }
}

// ---------------------------------------------------------------------------
extern "C" void kernel_launch(void* const* d_in, const int* in_sizes, int n_in,
                              void* d_out, int out_size, void* d_ws, size_t ws_size,
                              hipStream_t stream) {
    const float* codes  = (const float*)d_in[0];   // [16,128,128,128]
    const int*   segmap = (const int*)  d_in[1];   // [16,4,1,256,256]
    const float* fcw    = (const float*)d_in[2];   // [512,128]
    const float* fcb    = (const float*)d_in[3];   // [512]
    float*       outp   = (float*)d_out;           // [16,4,512]

    char* ws = (char*)d_ws;
    int*           counts = (int*)ws;                              // 64 * 4 B
    float*         sums   = (float*)(ws + 256);                    // 64*128*4 B
    unsigned char* mask   = (unsigned char*)(ws + 256 + MROWS * NF * 4); // 256 KiB

    (void)hipMemsetAsync(counts, 0, MROWS * sizeof(int), stream);
    seg_mask_kernel<<<NB * 8, 256, 0, stream>>>(segmap, mask, counts);
    masked_sum_kernel<<<NB * NF, 256, 0, stream>>>(codes, mask, sums);
    feats_gemm_wmma<<<16, 256, 0, stream>>>(sums, counts, fcw, fcb, outp);
}